// HeatSolver_15204184228514
// MI455X (gfx1250) — compile-verified
//
#include <hip/hip_runtime.h>
#include <hip/hip_bf16.h>

// ---------------------------------------------------------------------------
// Heat equation: 400 sequential explicit steps on 512x512, full history out.
// Roofline: ~420 MB compulsory history writes -> ~18us @ 23.3 TB/s. With one
// kernel per step, graph-node overhead (~400 x 1-2us) dominates 20x, so we
// fuse S=8 steps per launch via temporal blocking: stage a (TX+2S)x(TY+2S)
// tile of u & sigma into LDS (CDNA5 async global->LDS, ASYNCcnt), run 8
// sub-steps in LDS ping-pong, write each sub-step's exact interior to its
// history slice. Correct region erodes 1 cell/side/sub-step; halo=S keeps the
// interior exact. Clamped indexing == edge-replicated Neumann BC.
// Source term factored: f = sin(pi*x_i)*sin(pi*x_j)*exp(-t) -> 1-D LDS tables
// + one expf per sub-step (per-cell transcendentals would cost ~75us VALU).
// ---------------------------------------------------------------------------

#define S_SUB 8
#define TX 64
#define TY 32
#define EX (TX + 2 * S_SUB)   // 80
#define EY (TY + 2 * S_SUB)   // 48
#define NTHREADS 1024
#define PI_F 3.14159265358979323846f

// gfx1250 async global->LDS support (guarded so compile never hard-fails)
#if defined(__gfx1250__) && __has_builtin(__builtin_amdgcn_global_load_async_to_lds_b32)
#define HAS_ASYNC_LDS 1
#else
#define HAS_ASYNC_LDS 0
#endif

#if defined(__gfx1250__) && __has_builtin(__builtin_amdgcn_s_wait_asynccnt)
#define WAIT_ASYNC() __builtin_amdgcn_s_wait_asynccnt(0)
#elif HAS_ASYNC_LDS
#define WAIT_ASYNC() asm volatile("s_wait_asynccnt 0" ::: "memory")
#else
#define WAIT_ASYNC() ((void)0)
#endif

#if HAS_ASYNC_LDS
// Builtin signature (from clang diagnostic): (as1 int*, as3 int*, imm, imm)
typedef __attribute__((address_space(1))) int g1_int;
typedef __attribute__((address_space(3))) int l3_int;
#endif

// S_SUB fused explicit-Euler steps with redundant-halo temporal blocking.
__global__ __launch_bounds__(NTHREADS) void heat_multistep_kernel(
    const float* __restrict__ sigma,
    const float* __restrict__ Tptr,
    float*       __restrict__ hist,   // u_history base (slice 0 .. n_steps)
    int Mg, int k0, int steps_this, int n_steps)
{
    __shared__ float uA[EY * EX];
    __shared__ float uB[EY * EX];
    __shared__ float sg[EY * EX];
    __shared__ float sx[EX];          // sin(pi*x_col)
    __shared__ float sy[EY];          // sin(pi*x_row)

    const int tid = threadIdx.x;
    const int ox  = blockIdx.x * TX;  // interior origin, global col (j)
    const int oy  = blockIdx.y * TY;  // interior origin, global row (i)
    const size_t MM = (size_t)Mg * Mg;
    const float* u_in = hist + (size_t)k0 * MM;

    // ---- stage extended u and sigma tiles with clamped halo (async) ----
    for (int idx = tid; idx < EX * EY; idx += NTHREADS) {
        const int lx = idx % EX;
        const int ly = idx / EX;
        int gxx = ox + lx - S_SUB; gxx = min(max(gxx, 0), Mg - 1);
        int gyy = oy + ly - S_SUB; gyy = min(max(gyy, 0), Mg - 1);
        const size_t off = (size_t)gyy * Mg + gxx;
#if HAS_ASYNC_LDS
        __builtin_amdgcn_global_load_async_to_lds_b32(
            (g1_int*)(u_in + off), (l3_int*)&uA[idx], 0, 0);
        __builtin_amdgcn_global_load_async_to_lds_b32(
            (g1_int*)(sigma + off), (l3_int*)&sg[idx], 0, 0);
#else
        uA[idx] = u_in[off];
        sg[idx] = sigma[off];
#endif
    }

    // ---- separable source factors (one sinf per row/col of the tile) ----
    const float h = 1.0f / (float)Mg;
    if (tid < EX) {
        const int g = min(max(ox + tid - S_SUB, 0), Mg - 1);
        sx[tid] = sinf(PI_F * (((float)g + 0.5f) * h));
    } else if (tid >= 128 && tid < 128 + EY) {
        const int g = min(max(oy + (tid - 128) - S_SUB, 0), Mg - 1);
        sy[tid - 128] = sinf(PI_F * (((float)g + 0.5f) * h));
    }

    WAIT_ASYNC();          // drain this wave's ASYNCcnt before the barrier
    __syncthreads();       // all waves' LDS writes now visible

    const float Tv     = Tptr[0];
    const float tau    = Tv / (float)n_steps;
    const float inv_h2 = (float)Mg * (float)Mg;

    float* cur = uA;
    float* nxt = uB;
    for (int s = 0; s < steps_this; ++s) {
        const float e_t = expf(-((float)(k0 + s) * tau));
        float* u_out = hist + (size_t)(k0 + s + 1) * MM;
        const int lo  = s + 1;              // correct region after s+1 steps
        const int hix = EX - 2 - s;
        const int hiy = EY - 2 - s;

        for (int idx = tid; idx < EX * EY; idx += NTHREADS) {
            const int lx = idx % EX;
            const int ly = idx / EX;
            if (lx < lo || lx > hix || ly < lo || ly > hiy) continue;
            const int gxx = ox + lx - S_SUB;
            const int gyy = oy + ly - S_SUB;
            if (gxx < 0 || gxx >= Mg || gyy < 0 || gyy >= Mg) continue;

            // clamped neighbors == edge-replicated ghost cells
            const int lxm = (gxx > 0)      ? lx - 1 : lx;
            const int lxp = (gxx < Mg - 1) ? lx + 1 : lx;
            const int lym = (gyy > 0)      ? ly - 1 : ly;
            const int lyp = (gyy < Mg - 1) ? ly + 1 : ly;

            const float uc  = cur[idx];
            const float lap = cur[lyp * EX + lx] + cur[lym * EX + lx]
                            + cur[ly * EX + lxp] + cur[ly * EX + lxm]
                            - 4.0f * uc;
            const float f   = (sy[ly] * sx[lx]) * e_t;
            const float un  = uc + tau * (sg[idx] * lap * inv_h2 + f);

            nxt[idx] = un;
            // interior cells own the history write for this sub-step
            if (lx >= S_SUB && lx < S_SUB + TX && ly >= S_SUB && ly < S_SUB + TY)
                u_out[(size_t)gyy * Mg + gxx] = un;
        }
        __syncthreads();
        float* t = cur; cur = nxt; nxt = t;
    }
}

__global__ __launch_bounds__(NTHREADS) void heat_zero_kernel(float4* __restrict__ p, int n4) {
    int i = blockIdx.x * blockDim.x + threadIdx.x;
    if (i < n4) p[i] = make_float4(0.f, 0.f, 0.f, 0.f);
}

extern "C" void kernel_launch(void* const* d_in, const int* in_sizes, int n_in,
                              void* d_out, int out_size, void* d_ws, size_t ws_size,
                              hipStream_t stream) {
    const float* sigma = (const float*)d_in[0];
    const float* Tptr  = (const float*)d_in[1];
    (void)n_in; (void)d_ws; (void)ws_size;

    // Grid size from sigma's element count (integer sqrt).
    int Mg = 1;
    while ((long long)(Mg + 1) * (Mg + 1) <= (long long)in_sizes[0]) ++Mg;
    const int MM = Mg * Mg;

    // out = [u (MM)] ++ [u_history ((n_steps+1)*MM)]  ->  n_steps from out_size
    const int n_steps = out_size / MM - 2;

    float* u_final = (float*)d_out;        // first output: final u
    float* hist    = (float*)d_out + MM;   // second output: u_history[0..n_steps]

    // slice 0 = u0 = zeros
    {
        const int n4 = MM / 4;
        heat_zero_kernel<<<(n4 + NTHREADS - 1) / NTHREADS, NTHREADS, 0, stream>>>(
            (float4*)hist, n4);
    }

    // fused chain: 8 steps per node -> ~52 graph nodes instead of ~402
    dim3 grid((Mg + TX - 1) / TX, (Mg + TY - 1) / TY);
    for (int k0 = 0; k0 < n_steps; k0 += S_SUB) {
        int st = n_steps - k0;
        if (st > S_SUB) st = S_SUB;
        heat_multistep_kernel<<<grid, NTHREADS, 0, stream>>>(
            sigma, Tptr, hist, Mg, k0, st, n_steps);
    }

    // u = last history slice
    (void)hipMemcpyAsync(u_final, hist + (size_t)n_steps * MM,
                         (size_t)MM * sizeof(float),
                         hipMemcpyDeviceToDevice, stream);
}